// SpatialTransformer3D_6975026888725
// MI455X (gfx1250) — compile-verified
//
#include <hip/hip_runtime.h>
#include <hip/hip_bf16.h>
#include <cmath>

typedef __attribute__((ext_vector_type(2))) float v2f;
typedef __attribute__((ext_vector_type(8))) float v8f;

#define NB    8
#define WW    144
#define HH    144
#define DD    144
#define WH    (WW * HH)
#define NVOX  (WW * HH * DD)        // 2,985,984 (divisible by 256)
#define BPB   (NVOX / 256)          // 11664 blocks per batch
#define EDGE  143.0f
#define CSCL  71.5f                 // (W-1)/2, same for all axes (cube, [-1,1])
#define GS    (2.0f / 143.0f)       // grid step in normalized coords

// Monotone uint encoding of float: order-preserving, so uint atomicMin == float min.
__device__ __forceinline__ unsigned encf(float f) {
    unsigned u = __float_as_uint(f);
    return (u & 0x80000000u) ? ~u : (u | 0x80000000u);
}
__device__ __forceinline__ float decf(unsigned e) {
    return (e & 0x80000000u) ? __uint_as_float(e ^ 0x80000000u)
                             : __uint_as_float(~e);
}

__global__ void k_init(unsigned* ws) { ws[0] = 0xFF800000u; /* encf(+inf) */ }

// Doubles as the L2 warm-up pass for the image (default RT temporal hint).
__global__ __launch_bounds__(256) void k_min(const float* __restrict__ img, int n,
                                             unsigned* __restrict__ ws) {
    __shared__ unsigned sm[256];
    float m = INFINITY;
    for (int i = blockIdx.x * 256 + threadIdx.x; i < n; i += gridDim.x * 256)
        m = fminf(m, img[i]);
    sm[threadIdx.x] = encf(m);
    __syncthreads();
    for (int s = 128; s > 0; s >>= 1) {
        if (threadIdx.x < s) sm[threadIdx.x] = min(sm[threadIdx.x], sm[threadIdx.x + s]);
        __syncthreads();
    }
    if (threadIdx.x == 0) atomicMin(ws, sm[0]);
}

__global__ __launch_bounds__(256) void k_st3d(const float* __restrict__ img,
                                              const float* __restrict__ tr,
                                              float* __restrict__ out,
                                              const unsigned* __restrict__ ws) {
    const int tid   = threadIdx.x;
    const int lane  = tid & 31;          // wave32
    const int m16   = lane & 15;
    const int half  = lane >> 4;         // 0: K=0,1 / rows 0..7 ; 1: K=2,3 / rows 8..15

    const int blk    = blockIdx.x;
    const int b      = blk / BPB;
    const int blkInB = blk - b * BPB;
    const int base   = blkInB * 256 + (tid >> 5) * 32;   // within-batch voxel id, lane 0
    const int vown   = base + lane;                      // this lane's voxel

    // ---- scaled theta: ix = t00*gx + t01*gy + t02*gz + t03, coords in voxel units
    const float* q = tr + b * 7;
    const float qw = q[0], qx = q[1], qy = q[2], qz = q[3];
    const float t00 = CSCL * (1.0f - 2.0f * (qy * qy + qz * qz));
    const float t01 = CSCL * (2.0f * (qx * qy - qw * qz));
    const float t02 = CSCL * (2.0f * (qw * qy + qx * qz));
    const float t03 = CSCL * (q[4] + 1.0f);
    const float t10 = CSCL * (2.0f * (qx * qy + qw * qz));
    const float t11 = CSCL * (1.0f - 2.0f * (qx * qx + qz * qz));
    const float t12 = CSCL * (2.0f * (qy * qz - qw * qx));
    const float t13 = CSCL * (q[5] + 1.0f);
    const float t20 = CSCL * (2.0f * (qx * qz - qw * qy));
    const float t21 = CSCL * (2.0f * (qw * qx + qy * qz));
    const float t22 = CSCL * (1.0f - 2.0f * (qx * qx + qy * qy));
    const float t23 = CSCL * (q[6] + 1.0f);

    // ---- A matrix (16x4 f32): theta rows at M=0..2 and duplicated at M=8..10.
    // Lane L: M = L&15, VGPR0 = K=(half?2:0), VGPR1 = K=(half?3:1).
    const int r = (m16 < 3) ? m16 : ((m16 >= 8 && m16 < 11) ? (m16 - 8) : -1);
    const float cA = (r == 0) ? t00 : (r == 1) ? t10 : t20;
    const float cB = (r == 0) ? t01 : (r == 1) ? t11 : t21;
    const float cC = (r == 0) ? t02 : (r == 1) ? t12 : t22;
    const float cD = (r == 0) ? t03 : (r == 1) ? t13 : t23;
    v2f A;
    A.x = (r < 0) ? 0.0f : (half ? cC : cA);
    A.y = (r < 0) ? 0.0f : (half ? cD : cB);

    // ---- B matrices (4x16 f32): column n = homogeneous grid point of one voxel.
    // Lane L: N = L&15, VGPR0 = K=(half?2:0) -> (half? gz : gx), VGPR1 = (half? 1 : gy).
    const int u1 = base + m16;        // voxel group 0 (lanes 0..15 own these)
    const int u2 = base + 16 + m16;   // voxel group 1 (lanes 16..31 own these)

    int xg = u1 % WW; int tt = u1 / WW; int yg = tt % HH; int zg = tt / HH;
    v2f B1;
    B1.x = half ? (zg * GS - 1.0f) : (xg * GS - 1.0f);
    B1.y = half ? 1.0f             : (yg * GS - 1.0f);

    xg = u2 % WW; tt = u2 / WW; yg = tt % HH; zg = tt / HH;
    v2f B2;
    B2.x = half ? (zg * GS - 1.0f) : (xg * GS - 1.0f);
    B2.y = half ? 1.0f             : (yg * GS - 1.0f);

    // ---- coordinate transform on the matrix pipe (EXEC is all-1s here: no
    // divergent branches above, only cndmask selects).
    v8f cz = {};
    v8f acc1 = __builtin_amdgcn_wmma_f32_16x16x4_f32(false, A, false, B1,
                                                     (short)0, cz, false, false);
    v8f acc2 = __builtin_amdgcn_wmma_f32_16x16x4_f32(false, A, false, B2,
                                                     (short)0, cz, false, false);

    // Lanes 0..15 read rows 0..2 of acc1; lanes 16..31 read rows 8..10 of acc2
    // (both land in accumulator VGPRs 0..2 for the respective half-wave).
    const float ix = half ? acc2[0] : acc1[0];
    const float iy = half ? acc2[1] : acc1[1];
    const float iz = half ? acc2[2] : acc1[2];

    // ---- trilinear gather with border clamp (matches reference semantics)
    const bool inb = (ix >= 0.0f) & (ix <= EDGE) & (iy >= 0.0f) & (iy <= EDGE) &
                     (iz >= 0.0f) & (iz <= EDGE);

    float ixf = fminf(fmaxf(floorf(ix), 0.0f), EDGE);
    float iyf = fminf(fmaxf(floorf(iy), 0.0f), EDGE);
    float izf = fminf(fmaxf(floorf(iz), 0.0f), EDGE);
    const float fx = ix - ixf, fy = iy - iyf, fz = iz - izf;
    const int x0 = (int)ixf, y0 = (int)iyf, z0 = (int)izf;
    const int x1 = min(x0 + 1, WW - 1);
    const int y1 = min(y0 + 1, HH - 1);
    const int z1 = min(z0 + 1, DD - 1);

    const float* p = img + (size_t)b * NVOX;
    const int zy00 = z0 * WH + y0 * WW;
    const int zy01 = z0 * WH + y1 * WW;
    const int zy10 = z1 * WH + y0 * WW;
    const int zy11 = z1 * WH + y1 * WW;

    const float v000 = p[zy00 + x0], v100 = p[zy00 + x1];
    const float v010 = p[zy01 + x0], v110 = p[zy01 + x1];
    const float v001 = p[zy10 + x0], v101 = p[zy10 + x1];
    const float v011 = p[zy11 + x0], v111 = p[zy11 + x1];

    const float wx0 = 1.0f - fx, wy0 = 1.0f - fy, wz0 = 1.0f - fz;
    const float r0 = wy0 * fmaf(wx0, v000, fx * v100) + fy * fmaf(wx0, v010, fx * v110);
    const float r1 = wy0 * fmaf(wx0, v001, fx * v101) + fy * fmaf(wx0, v011, fx * v111);
    const float res = fmaf(wz0, r0, fz * r1);

    const float pad = decf(ws[0]);
    // Non-temporal store: output is write-once / never re-read; keep the image
    // (95.5 MB) resident in the 192 MB L2 instead of letting output evict it.
    __builtin_nontemporal_store(inb ? res : pad, &out[(size_t)b * NVOX + vown]);
}

extern "C" void kernel_launch(void* const* d_in, const int* in_sizes, int n_in,
                              void* d_out, int out_size, void* d_ws, size_t ws_size,
                              hipStream_t stream) {
    const float* img = (const float*)d_in[0];   // [8,144,144,144,1] f32
    const float* trf = (const float*)d_in[1];   // [8,7] f32 (quat + translation)
    float* out       = (float*)d_out;
    unsigned* wsm    = (unsigned*)d_ws;
    const int nTot   = in_sizes[0];             // 23,887,872

    k_init<<<1, 1, 0, stream>>>(wsm);
    k_min<<<2048, 256, 0, stream>>>(img, nTot, wsm);
    k_st3d<<<NB * BPB, 256, 0, stream>>>(img, trf, out, wsm);
}